// DrugEncoder_37409165148770
// MI455X (gfx1250) — compile-verified
//
#include <hip/hip_runtime.h>

typedef float v2f __attribute__((ext_vector_type(2)));
typedef float v8f __attribute__((ext_vector_type(8)));

// ---------------------------------------------------------------------------
// Degree / normalization kernels (GCN symmetric norm with self-loops)
// ---------------------------------------------------------------------------
__global__ void init_deg_kernel(float* __restrict__ deg, int n) {
    int i = blockIdx.x * blockDim.x + threadIdx.x;
    if (i < n) deg[i] = 1.0f;  // self-loop contributes 1
}

__global__ void count_deg_kernel(const int* __restrict__ dst,
                                 float* __restrict__ deg, int E) {
    int e = blockIdx.x * blockDim.x + threadIdx.x;
    if (e < E) atomicAdd(&deg[dst[e]], 1.0f);
}

__global__ void dinv_kernel(const float* __restrict__ deg,
                            float* __restrict__ dinv, int n) {
    int i = blockIdx.x * blockDim.x + threadIdx.x;
    if (i < n) dinv[i] = rsqrtf(fmaxf(deg[i], 1.0f));
}

// ---------------------------------------------------------------------------
// GEMM: H[n_rows x Fout] = X[n_rows x K] * W[K x Fout]
// Full-precision V_WMMA_F32_16X16X4_F32. One wave owns a 16x64 output stripe
// (4 C accumulators sharing each A fragment). Requires K % 4 == 0,
// n_rows % 16 == 0, Fout % 64 == 0 (true here: K in {128,256},
// n_rows = 50000, Fout in {128,256,512}).
// ---------------------------------------------------------------------------
__global__ __launch_bounds__(128)
void gemm_wmma_f32_kernel(const float* __restrict__ X,
                          const float* __restrict__ W,
                          float* __restrict__ H,
                          int nRowTiles, int K, int Fout) {
    const int lane = threadIdx.x & 31;
    const int wid  = threadIdx.x >> 5;
    const int gw   = blockIdx.x * 4 + wid;

    const int colGroups = Fout >> 6;          // 64 columns per wave
    const int rowTile   = gw / colGroups;
    if (rowTile >= nRowTiles) return;         // wave-uniform guard (EXEC all-1s)
    const int colBase   = (gw % colGroups) << 6;

    const int m  = lane & 15;                 // A row / C-D col index
    const int hi = lane >> 4;                 // half-wave selector

    const float* xrow = X + (size_t)(rowTile * 16 + m) * K;

    v8f acc0 = {}, acc1 = {}, acc2 = {}, acc3 = {};

    for (int k = 0; k < K; k += 4) {
        // A fragment: 16x4, lane holds K = k+2*hi .. k+2*hi+1 of row m
        v2f a;
        a.x = xrow[k + 2 * hi];
        a.y = xrow[k + 2 * hi + 1];

        // B fragments: 4x16, lane holds rows k+2*hi, k+2*hi+1 at col (lane&15)
        const float* wr0 = W + (size_t)(k + 2 * hi) * Fout + colBase + m;
        const float* wr1 = wr0 + Fout;
        v2f b0; b0.x = wr0[0];  b0.y = wr1[0];
        v2f b1; b1.x = wr0[16]; b1.y = wr1[16];
        v2f b2; b2.x = wr0[32]; b2.y = wr1[32];
        v2f b3; b3.x = wr0[48]; b3.y = wr1[48];

        acc0 = __builtin_amdgcn_wmma_f32_16x16x4_f32(false, a, false, b0,
                                                     (short)0, acc0, false, false);
        acc1 = __builtin_amdgcn_wmma_f32_16x16x4_f32(false, a, false, b1,
                                                     (short)0, acc1, false, false);
        acc2 = __builtin_amdgcn_wmma_f32_16x16x4_f32(false, a, false, b2,
                                                     (short)0, acc2, false, false);
        acc3 = __builtin_amdgcn_wmma_f32_16x16x4_f32(false, a, false, b3,
                                                     (short)0, acc3, false, false);
    }

    // C/D layout: VGPR r -> row (r + 8*hi), col (lane&15)
    float* hbase = H + (size_t)(rowTile * 16) * Fout + colBase + m;
#pragma unroll
    for (int r = 0; r < 8; ++r) {
        const size_t roff = (size_t)(r + 8 * hi) * Fout;
        hbase[roff +  0] = acc0[r];
        hbase[roff + 16] = acc1[r];
        hbase[roff + 32] = acc2[r];
        hbase[roff + 48] = acc3[r];
    }
}

// ---------------------------------------------------------------------------
// Edge scatter: agg[dst] += dinv[src]*dinv[dst] * h[src]
// One wave per edge per 128-float feature chunk; scalar edge metadata,
// lane-level float4 gather + 4 global f32 atomics.
// ---------------------------------------------------------------------------
__global__ __launch_bounds__(128)
void edge_agg_kernel(const int* __restrict__ src, const int* __restrict__ dst,
                     const float* __restrict__ dinv,
                     const float* __restrict__ h, float* __restrict__ agg,
                     int E, int F) {
    const int lane = threadIdx.x & 31;
    const int wid  = threadIdx.x >> 5;
    const int e    = blockIdx.x * 4 + wid;
    if (e >= E) return;

    const int s = src[e];
    const int d = dst[e];
    const float c = dinv[s] * dinv[d];

    const int f = blockIdx.y * 128 + lane * 4;
    const float4 hv = *reinterpret_cast<const float4*>(h + (size_t)s * F + f);
    float* out = agg + (size_t)d * F + f;
    atomicAdd(out + 0, c * hv.x);
    atomicAdd(out + 1, c * hv.y);
    atomicAdd(out + 2, c * hv.z);
    atomicAdd(out + 3, c * hv.w);
}

// ---------------------------------------------------------------------------
// Epilogue: out = relu(agg + dinv[i]^2 * h[i,:] + bias)  (self-loop folded in)
// fShift = log2(F) so node/feature split is shift/mask.
// ---------------------------------------------------------------------------
__global__ __launch_bounds__(256)
void epilogue_kernel(const float* __restrict__ agg,
                     const float* __restrict__ h,
                     const float* __restrict__ dinv,
                     const float* __restrict__ bias,
                     float* __restrict__ out,
                     long long total, int fShift, int fMask) {
    long long idx4 = ((long long)blockIdx.x * blockDim.x + threadIdx.x) * 4;
    if (idx4 >= total) return;
    const int i = (int)(idx4 >> fShift);
    const int f = (int)(idx4 & fMask);
    float w = dinv[i];
    w = w * w;
    const float4 a  = *reinterpret_cast<const float4*>(agg + idx4);
    const float4 hv = *reinterpret_cast<const float4*>(h + idx4);
    const float4 b  = *reinterpret_cast<const float4*>(bias + f);
    float4 r;
    r.x = fmaxf(fmaf(w, hv.x, a.x) + b.x, 0.0f);
    r.y = fmaxf(fmaf(w, hv.y, a.y) + b.y, 0.0f);
    r.z = fmaxf(fmaf(w, hv.z, a.z) + b.z, 0.0f);
    r.w = fmaxf(fmaf(w, hv.w, a.w) + b.w, 0.0f);
    *reinterpret_cast<float4*>(out + idx4) = r;
}

// ---------------------------------------------------------------------------
// Launch
// ---------------------------------------------------------------------------
static void run_layer(const float* xin, int K, const float* Wm, const float* bv,
                      int Fout, float* h, float* xout,
                      const int* src, const int* dst, const float* dinv,
                      int N, int E, hipStream_t stream) {
    const int nRowTiles = N / 16;                 // 50000/16 = 3125 exact
    const int waves = nRowTiles * (Fout / 64);
    gemm_wmma_f32_kernel<<<(waves + 3) / 4, 128, 0, stream>>>(
        xin, Wm, h, nRowTiles, K, Fout);

    hipMemsetAsync(xout, 0, (size_t)N * Fout * sizeof(float), stream);

    dim3 eg((E + 3) / 4, Fout / 128);
    edge_agg_kernel<<<eg, 128, 0, stream>>>(src, dst, dinv, h, xout, E, Fout);

    const long long total = (long long)N * Fout;
    int fShift = 0;
    while ((1 << fShift) < Fout) ++fShift;
    epilogue_kernel<<<(unsigned)((total / 4 + 255) / 256), 256, 0, stream>>>(
        xout, h, dinv, bv, xout, total, fShift, Fout - 1);
}

extern "C" void kernel_launch(void* const* d_in, const int* in_sizes, int n_in,
                              void* d_out, int out_size, void* d_ws, size_t ws_size,
                              hipStream_t stream) {
    const float* v   = (const float*)d_in[0];
    const int*   ei  = (const int*)d_in[1];
    const float* W1  = (const float*)d_in[2];
    const float* b1  = (const float*)d_in[3];
    const float* W2  = (const float*)d_in[4];
    const float* b2  = (const float*)d_in[5];
    const float* W3  = (const float*)d_in[6];
    const float* b3  = (const float*)d_in[7];

    const int N = in_sizes[0] / 128;   // 50000
    const int E = in_sizes[1] / 2;     // 400000
    const int* src = ei;
    const int* dst = ei + E;

    float* ws   = (float*)d_ws;
    float* deg  = ws;
    float* dinv = deg  + N;
    float* x1   = dinv + N;
    float* x2   = x1 + (size_t)N * 128;
    float* h    = x2 + (size_t)N * 256;   // sized for max Fout = 512
    float* out  = (float*)d_out;

    // Symmetric normalization coefficients (computed once per call)
    init_deg_kernel<<<(N + 255) / 256, 256, 0, stream>>>(deg, N);
    count_deg_kernel<<<(E + 255) / 256, 256, 0, stream>>>(dst, deg, E);
    dinv_kernel<<<(N + 255) / 256, 256, 0, stream>>>(deg, dinv, N);

    // Three GCN layers
    run_layer(v,  128, W1, b1, 128, h, x1,  src, dst, dinv, N, E, stream);
    run_layer(x1, 128, W2, b2, 256, h, x2,  src, dst, dinv, N, E, stream);
    run_layer(x2, 256, W3, b3, 512, h, out, src, dst, dinv, N, E, stream);
}